// NNFFTLayer_78941498901166
// MI455X (gfx1250) — compile-verified
//
#include <hip/hip_runtime.h>

typedef __attribute__((ext_vector_type(2))) float v2f;
typedef __attribute__((ext_vector_type(8))) float v8f;

#define N_TOT  8192   // input_size
#define TWO_D  1024   // 2*dilation = len(weights), scale period (floats)
#define RADIX  8

// Each wave processes 8 tiles of 256 floats (one row covered by 4 waves).
// Tiles are strided by 4 (=1024 floats = the scale period), so the per-lane
// scale values, the A operand (diag(lin,lin)) and the store-offset math are
// all loop-invariant. Per tile: D(16x16) = diag(lin,lin) * B(16x16) as
// 4 chained V_WMMA_F32_16X16X4_F32.
__global__ __launch_bounds__(256) void nnfft_wmma_kernel(
    const float* __restrict__ x,
    const float* __restrict__ w,
    const float* __restrict__ lin,
    float* __restrict__ out,
    int rows)
{
    const int lane      = threadIdx.x & 31;
    const int waveInBlk = threadIdx.x >> 5;
    const int wave      = blockIdx.x * 8 + waveInBlk;   // 8 waves per block
    const int row       = wave >> 2;                    // 4 waves per 8192-float row
    const int g         = wave & 3;                     // tile-group: kpair = g + 4*j
    if (row >= rows) return;                            // wave-uniform (EXEC stays all-1s)

    const long base = (long)row * N_TOT + (long)g * 256;

    // Lane offsets inside a 256-float tile, per ISA WMMA VGPR striping:
    //   B VGPR pair for step t: rows {4t,4t+2} / {4t+1,4t+3}, cols = lane half.
    const int lo_e = (lane < 16) ? lane : (lane + 16);
    const int lo_o = lo_e + 16;
    //   D VGPR r holds rows {r, r+8} -> linear tile offset r*16 + so.
    const int so   = (lane < 16) ? lane : (lane + 112);

    // ---- A operand (loop-invariant): lin embedded block-diagonally in 16x16,
    //      split into 4 K=4 slices. Slice t VGPR0: A[m, 4t + (lane>=16?2:0)].
    const int m = lane & 15;
    v2f a[4];
#pragma unroll
    for (int t = 0; t < 4; ++t) {
        const int kc0 = 4 * t + ((lane >= 16) ? 2 : 0);
#pragma unroll
        for (int v = 0; v < 2; ++v) {
            const int  kc    = kc0 + v;
            const bool valid = ((m >= 8) == (kc >= 8));   // block-diagonal mask
            a[t][v] = valid ? lin[(m & 7) * 8 + (kc & 7)] : 0.0f;
        }
    }

    // ---- Scale values (loop-invariant): tile offset mod 1024 == g*256 + tile off.
    //      8 loads from the 4KB, L2-resident weights array, held in VGPRs.
    const int wbase = g * 256;
    float ws[8];
#pragma unroll
    for (int t = 0; t < 4; ++t) {
        ws[2 * t]     = w[wbase + 64 * t + lo_e];
        ws[2 * t + 1] = w[wbase + 64 * t + lo_o];
    }

    const float* __restrict__ xb = x   + base;
    float*       __restrict__ ob = out + base;

    // ---- Streaming loop: 8 tiles, stride 1024 floats (4096 B imm offsets). ----
#pragma unroll 2
    for (int j = 0; j < 8; ++j) {
        const int toff = j * TWO_D;

        // 8 coalesced NT b32 loads covering the contiguous 1KB tile, scaled.
        float b[8];
#pragma unroll
        for (int t = 0; t < 4; ++t) {
            b[2 * t]     = __builtin_nontemporal_load(&xb[toff + 64 * t + lo_e]) * ws[2 * t];
            b[2 * t + 1] = __builtin_nontemporal_load(&xb[toff + 64 * t + lo_o]) * ws[2 * t + 1];
        }

        // 4 chained K=4 fp32 WMMAs: acc = A_t x B_t + acc.
        v8f acc = {0.f, 0.f, 0.f, 0.f, 0.f, 0.f, 0.f, 0.f};
#pragma unroll
        for (int t = 0; t < 4; ++t) {
            v2f bb;
            bb[0] = b[2 * t];
            bb[1] = b[2 * t + 1];
            acc = __builtin_amdgcn_wmma_f32_16x16x4_f32(
                /*neg_a=*/false, a[t], /*neg_b=*/false, bb,
                /*c_mod=*/(short)0, acc, /*reuse_a=*/false, /*reuse_b=*/false);
        }

        // 8 coalesced NT b32 stores, same contiguous 1KB tile layout.
#pragma unroll
        for (int r = 0; r < 8; ++r)
            __builtin_nontemporal_store(acc[r], &ob[toff + so + 16 * r]);
    }
}

extern "C" void kernel_launch(void* const* d_in, const int* in_sizes, int n_in,
                              void* d_out, int out_size, void* d_ws, size_t ws_size,
                              hipStream_t stream) {
    const float* x   = (const float*)d_in[0];   // [BATCH, SEQ, N] fp32
    const float* w   = (const float*)d_in[1];   // [1024] fp32
    const float* lin = (const float*)d_in[2];   // [8, 8] fp32
    float*       out = (float*)d_out;

    const int rows   = in_sizes[0] / N_TOT;     // BATCH*SEQ (even)
    const int blocks = (rows * 4 + 7) / 8;      // 4 waves/row, 8 waves/block

    nnfft_wmma_kernel<<<blocks, 256, 0, stream>>>(x, w, lin, out, rows);
}